// M1_40441412059705
// MI455X (gfx1250) — compile-verified
//
#include <hip/hip_runtime.h>
#include <math.h>

typedef __attribute__((ext_vector_type(2))) float v2f;
typedef __attribute__((ext_vector_type(8))) float v8f;

#define HF 128  // feature/hidden width (fixed by the model)

// ---------------- degree / normalization ----------------
__global__ void deg_count_kernel(const int* __restrict__ col, float* __restrict__ deg, int E) {
  int e = blockIdx.x * blockDim.x + threadIdx.x;
  if (e < E) atomicAdd(&deg[col[e]], 1.0f);
}

__global__ void dinv_kernel(float* deg, int N) {
  int i = blockIdx.x * blockDim.x + threadIdx.x;
  if (i < N) deg[i] = 1.0f / sqrtf(deg[i] + 1.0f);  // +1 = self loop
}

// ---------------- fp32 WMMA GEMM: out[N,128] = A[N,128] @ W[128,128] ----------------
__global__ __launch_bounds__(256) void gemm128_wmma(const float* __restrict__ A,
                                                    const float* __restrict__ W,
                                                    float* __restrict__ out, int N) {
  __shared__ __align__(16) float As[16 * 132];  // 16 rows, stride 132 (conflict-free)
  const int row0 = blockIdx.x * 16;

  // cooperative load of the 16x128 A tile (512 float4's, 2 per thread)
  for (int i = threadIdx.x; i < 512; i += 256) {
    int r = i >> 5, c4 = i & 31;
    float4 v = make_float4(0.f, 0.f, 0.f, 0.f);
    int gr = row0 + r;
    if (gr < N) v = ((const float4*)(A + (size_t)gr * HF))[c4];
    *(float4*)&As[r * 132 + c4 * 4] = v;
  }
  __syncthreads();

  const int lane = threadIdx.x & 31;
  const int wv   = threadIdx.x >> 5;   // 0..7: which 16 output columns
  const int mn   = lane & 15;          // M index (A) / N index (B)
  const int kb   = (lane >> 4) << 1;   // K sub-offset: lanes 0-15 -> 0, 16-31 -> 2
  const int col0 = wv * 16;

  v8f c = {};
#pragma unroll
  for (int k0 = 0; k0 < HF; k0 += 4) {
    int k = k0 + kb;
    v2f a, b;
    // A 16x4 layout: lane<16: K=k0,k0+1 ; lane>=16: K=k0+2,k0+3 (row = mn)
    a.x = As[mn * 132 + k];
    a.y = As[mn * 132 + k + 1];
    // B 4x16 layout mirrored: per lane column n=mn, rows k / k+1
    b.x = W[k * HF + col0 + mn];
    b.y = W[(k + 1) * HF + col0 + mn];
    c = __builtin_amdgcn_wmma_f32_16x16x4_f32(false, a, false, b, (short)0, c, false, false);
  }

  // C/D layout: VGPR r: lanes 0-15 -> (M=r, N=lane), lanes 16-31 -> (M=r+8, N=lane-16)
  const int rbase = row0 + ((lane >> 4) << 3);
#pragma unroll
  for (int r = 0; r < 8; ++r) {
    int gr = rbase + r;
    if (gr < N) out[(size_t)gr * HF + col0 + mn] = c[r];
  }
}

// ---------------- edge gather+scale+scatter-add (one wave per edge) ----------------
__global__ void edge_scatter_kernel(const float* __restrict__ g,
                                    const int* __restrict__ row,
                                    const int* __restrict__ col,
                                    const float* __restrict__ dinv,
                                    float* acc, int E) {
  int wid = (int)(((size_t)blockIdx.x * blockDim.x + threadIdx.x) >> 5);
  if (wid >= E) return;
  int lane = threadIdx.x & 31;
  int r = row[wid], c = col[wid];
  float nn = dinv[r] * dinv[c];
  float4 v = ((const float4*)(g + (size_t)r * HF))[lane];
  float* o = acc + (size_t)c * HF + lane * 4;
  atomicAdd(o + 0, v.x * nn);
  atomicAdd(o + 1, v.y * nn);
  atomicAdd(o + 2, v.z * nn);
  atomicAdd(o + 3, v.w * nn);
}

// ---------------- out = relu(acc + g*dinv^2 + bias)  (self-loop fused) ----------------
__global__ void self_bias_relu_kernel(const float* __restrict__ acc,
                                      const float* g,   // may alias out
                                      const float* __restrict__ dinv,
                                      const float* __restrict__ bias,
                                      float* out, int N) {
  size_t t = (size_t)blockIdx.x * blockDim.x + threadIdx.x;  // N*32 lanes, float4 each
  size_t node = t >> 5;
  int q = (int)(t & 31);
  if (node >= (size_t)N) return;
  float d = dinv[node];
  float sn = d * d;
  float4 a  = ((const float4*)(acc + node * HF))[q];
  float4 gv = ((const float4*)(g + node * HF))[q];
  float4 bv = ((const float4*)bias)[q];
  float4 r;
  r.x = fmaxf(fmaf(gv.x, sn, a.x) + bv.x, 0.f);
  r.y = fmaxf(fmaf(gv.y, sn, a.y) + bv.y, 0.f);
  r.z = fmaxf(fmaf(gv.z, sn, a.z) + bv.z, 0.f);
  r.w = fmaxf(fmaf(gv.w, sn, a.w) + bv.w, 0.f);
  ((float4*)(out + node * HF))[q] = r;
}

// ---------------- pooling ----------------
__global__ void pool_cnt_kernel(const int* __restrict__ batch, float* cnt, int N) {
  int i = blockIdx.x * blockDim.x + threadIdx.x;
  if (i < N) atomicAdd(&cnt[batch[i]], 1.0f);
}

// batch is sorted: accumulate runs in registers, atomic only on graph boundary
__global__ __launch_bounds__(128) void pool_sum_kernel(const float* __restrict__ h,
                                                       const int* __restrict__ batch,
                                                       float* sums, int N) {
  const int f = threadIdx.x;  // 0..127 feature
  int start = blockIdx.x * 256;
  if (start >= N) return;
  int end = min(start + 256, N);
  float acc = 0.f;
  int cur = batch[start];
  for (int i = start; i < end; ++i) {
    int b = batch[i];
    if (b != cur) {
      atomicAdd(&sums[cur * HF + f], acc);
      acc = 0.f;
      cur = b;
    }
    acc += h[(size_t)i * HF + f];
  }
  atomicAdd(&sums[cur * HF + f], acc);
}

// ---------------- head: mean / BN(batch stats) / MLP / softmax ----------------
__global__ __launch_bounds__(128) void head_kernel(const float* __restrict__ sums,
                                                   const float* __restrict__ cnt,
                                                   const float* __restrict__ gamma,
                                                   const float* __restrict__ beta,
                                                   const float* __restrict__ Wl1,
                                                   const float* __restrict__ bl1,
                                                   const float* __restrict__ Wl2,
                                                   const float* __restrict__ bl2,
                                                   float* __restrict__ out,
                                                   int B, int C) {
  __shared__ float xn[64 * HF];
  __shared__ float z1[64 * HF];
  const int f = threadIdx.x;  // 0..127

  for (int b = 0; b < B; ++b)
    xn[b * HF + f] = sums[b * HF + f] / fmaxf(cnt[b], 1.0f);
  __syncthreads();

  float mu = 0.f;
  for (int b = 0; b < B; ++b) mu += xn[b * HF + f];
  mu /= (float)B;
  float var = 0.f;
  for (int b = 0; b < B; ++b) { float d = xn[b * HF + f] - mu; var += d * d; }
  var /= (float)B;  // biased, matches jnp.var
  float inv = 1.0f / sqrtf(var + 1e-5f);
  float gm = gamma[f], bt = beta[f];
  for (int b = 0; b < B; ++b)
    xn[b * HF + f] = (xn[b * HF + f] - mu) * inv * gm + bt;
  __syncthreads();

  // z1 = xn @ Wl1 + bl1 ; thread f owns column f, unroll 4 rows
  float bv = bl1[f];
  for (int b = 0; b < B; b += 4) {
    float a0 = bv, a1 = bv, a2 = bv, a3 = bv;
    for (int k = 0; k < HF; ++k) {
      float w = Wl1[k * HF + f];
      a0 = fmaf(xn[(b + 0) * HF + k], w, a0);
      a1 = fmaf(xn[(b + 1) * HF + k], w, a1);
      a2 = fmaf(xn[(b + 2) * HF + k], w, a2);
      a3 = fmaf(xn[(b + 3) * HF + k], w, a3);
    }
    z1[(b + 0) * HF + f] = a0;
    z1[(b + 1) * HF + f] = a1;
    z1[(b + 2) * HF + f] = a2;
    z1[(b + 3) * HF + f] = a3;
  }
  __syncthreads();

  // z2 = z1 @ Wl2 + bl2, softmax per row; thread b handles row b
  if (f < B) {
    float z[16];
    float mx = -3.0e38f;
    for (int c = 0; c < C; ++c) {
      float a = bl2[c];
      for (int k = 0; k < HF; ++k) a = fmaf(z1[f * HF + k], Wl2[k * C + c], a);
      z[c] = a;
      mx = fmaxf(mx, a);
    }
    float s = 0.f;
    for (int c = 0; c < C; ++c) { z[c] = expf(z[c] - mx); s += z[c]; }
    float rs = 1.0f / s;
    for (int c = 0; c < C; ++c) out[f * C + c] = z[c] * rs;
  }
}

// ---------------- launch ----------------
extern "C" void kernel_launch(void* const* d_in, const int* in_sizes, int n_in,
                              void* d_out, int out_size, void* d_ws, size_t ws_size,
                              hipStream_t stream) {
  const float* x     = (const float*)d_in[0];
  const int*   ei    = (const int*)d_in[1];
  const int*   batch = (const int*)d_in[2];
  const float* W1 = (const float*)d_in[4];
  const float* b1 = (const float*)d_in[5];
  const float* W2 = (const float*)d_in[6];
  const float* b2 = (const float*)d_in[7];
  const float* W3 = (const float*)d_in[8];
  const float* b3 = (const float*)d_in[9];
  const float* gamma = (const float*)d_in[10];
  const float* beta  = (const float*)d_in[11];
  const float* Wl1 = (const float*)d_in[12];
  const float* bl1 = (const float*)d_in[13];
  const float* Wl2 = (const float*)d_in[14];
  const float* bl2 = (const float*)d_in[15];
  float* out = (float*)d_out;

  const int N = in_sizes[0] / HF;    // 50000
  const int E = in_sizes[1] / 2;     // 800000
  const int C = in_sizes[14] / HF;   // 10
  const int B = out_size / C;        // 64

  const int* erow = ei;
  const int* ecol = ei + E;

  // workspace layout
  float* P    = (float*)d_ws;               // N*HF
  float* Q    = P + (size_t)N * HF;         // N*HF
  float* dinv = Q + (size_t)N * HF;         // N
  float* sums = dinv + N;                   // B*HF
  float* cnt  = sums + (size_t)B * HF;      // B
  (void)ws_size; (void)n_in;

  const size_t hbytes = (size_t)N * HF * sizeof(float);
  const int gemmBlocks = (N + 15) / 16;
  const int scatBlocks = (E + 7) / 8;                 // 8 waves / block, 1 wave / edge
  const int nodeBlocks = (int)(((size_t)N * 32 + 255) / 256);

  // normalization factors
  hipMemsetAsync(dinv, 0, (size_t)N * sizeof(float), stream);
  deg_count_kernel<<<(E + 255) / 256, 256, 0, stream>>>(ecol, dinv, E);
  dinv_kernel<<<(N + 255) / 256, 256, 0, stream>>>(dinv, N);

  // layer 1: P = x@W1 ; Q = scatter(P) ; P = relu(Q + P*self + b1)
  gemm128_wmma<<<gemmBlocks, 256, 0, stream>>>(x, W1, P, N);
  hipMemsetAsync(Q, 0, hbytes, stream);
  edge_scatter_kernel<<<scatBlocks, 256, 0, stream>>>(P, erow, ecol, dinv, Q, E);
  self_bias_relu_kernel<<<nodeBlocks, 256, 0, stream>>>(Q, P, dinv, b1, P, N);

  // layer 2: Q = P@W2 ; P = scatter(Q) ; Q = relu(P + Q*self + b2)
  gemm128_wmma<<<gemmBlocks, 256, 0, stream>>>(P, W2, Q, N);
  hipMemsetAsync(P, 0, hbytes, stream);
  edge_scatter_kernel<<<scatBlocks, 256, 0, stream>>>(Q, erow, ecol, dinv, P, E);
  self_bias_relu_kernel<<<nodeBlocks, 256, 0, stream>>>(P, Q, dinv, b2, Q, N);

  // layer 3: P = Q@W3 ; Q = scatter(P) ; P = relu(Q + P*self + b3)
  gemm128_wmma<<<gemmBlocks, 256, 0, stream>>>(Q, W3, P, N);
  hipMemsetAsync(Q, 0, hbytes, stream);
  edge_scatter_kernel<<<scatBlocks, 256, 0, stream>>>(P, erow, ecol, dinv, Q, E);
  self_bias_relu_kernel<<<nodeBlocks, 256, 0, stream>>>(Q, P, dinv, b3, P, N);

  // pooling
  hipMemsetAsync(sums, 0, (size_t)B * HF * sizeof(float), stream);
  hipMemsetAsync(cnt, 0, (size_t)B * sizeof(float), stream);
  pool_cnt_kernel<<<(N + 255) / 256, 256, 0, stream>>>(batch, cnt, N);
  pool_sum_kernel<<<(N + 255) / 256, 128, 0, stream>>>(P, batch, sums, N);

  // BN + MLP + softmax
  head_kernel<<<1, 128, 0, stream>>>(sums, cnt, gamma, beta, Wl1, bl1, Wl2, bl2, out, B, C);
}